// Quantizer_20366734917960
// MI455X (gfx1250) — compile-verified
//
#include <hip/hip_runtime.h>
#include <hip/hip_bf16.h>

typedef __attribute__((ext_vector_type(2))) float v2f;
typedef __attribute__((ext_vector_type(8))) float v8f;

#define EMBED 512
#define VOCAB 512
#define XS_STRIDE 516   // 512 + 4 floats padding: conflict-free b64 LDS reads

// ---------------------------------------------------------------------------
// Kernel 1: bias[v] = sum_k W[k, v]^2   (column norms, per the reference quirk)
// ---------------------------------------------------------------------------
__global__ __launch_bounds__(256) void quantizer_bias_kernel(
    const float* __restrict__ w, float* __restrict__ bias) {
  const int v = blockIdx.x * blockDim.x + threadIdx.x;  // 0..511
  float s = 0.0f;
  #pragma unroll 8
  for (int k = 0; k < VOCAB; ++k) {
    const float x = w[(long)k * EMBED + v];
    s = fmaf(x, x, s);
  }
  bias[v] = s;
}

// ---------------------------------------------------------------------------
// Kernel 2: per 16-row tile -> f32 WMMA GEMM vs all 512 codes, argmin, gather
// ---------------------------------------------------------------------------
__global__ __launch_bounds__(256) void quantizer_main_kernel(
    const float* __restrict__ x, const float* __restrict__ w,
    const float* __restrict__ bias, float* __restrict__ out) {
  __shared__ float Xs[16 * XS_STRIDE];     // 16 rows x 512 (+pad) of X tile
  __shared__ float redV[8][16];            // per-wave per-row min value
  __shared__ int   redI[8][16];            // per-wave per-row argmin
  __shared__ int   fidx[16];               // final argmin per row

  const int tid  = threadIdx.x;
  // Wave index is identical across all 32 lanes -> force it scalar so the
  // compiler emits uniform (SALU) control flow, keeping EXEC all-1s around
  // every WMMA without lane masking.
  const int wave = __builtin_amdgcn_readfirstlane(tid >> 5);
  const int lane = tid & 31;
  const int lo   = lane & 15;              // A: row M source / B: column N source
  const int hi   = lane >> 4;              // selects K sub-pair within WMMA frag
  const long r0  = (long)blockIdx.x * 16;  // first row of this tile

  // ---- Stage X tile into LDS (float4, fully coalesced) --------------------
  #pragma unroll
  for (int i = 0; i < 8; ++i) {
    const int t   = tid + i * 256;         // 0..2047 float4 slots
    const int row = t >> 7;                // 128 float4 per row
    const int c4  = t & 127;
    const float4 val = ((const float4*)(x + (r0 + row) * EMBED))[c4];
    *(float4*)&Xs[row * XS_STRIDE + c4 * 4] = val;
  }
  __syncthreads();

  // ---- Per-lane running argmin for this lane's 8 rows ----------------------
  float best[8];
  int   bidx[8];
  #pragma unroll
  for (int j = 0; j < 8; ++j) { best[j] = __builtin_inff(); bidx[j] = 0; }

  const float* xrow = Xs + lo * XS_STRIDE + 2 * hi;

  // Each wave handles vocab tiles vt = wave + 8*t, t = 0..3 (v increasing).
  // Compile-time trip count: fully unrolled, uniform control flow.
  #pragma unroll
  for (int t = 0; t < 4; ++t) {
    const int vt = wave + t * 8;
    const int v  = vt * 16 + lo;                      // this lane's column
    const float* wrow = w + (long)v * EMBED + 2 * hi; // B frag source

    v8f c0 = {};
    v8f c1 = {};
    #pragma unroll 4
    for (int k0 = 0; k0 < EMBED; k0 += 8) {
      const v2f a0 = *(const v2f*)(xrow + k0);
      const v2f b0 = *(const v2f*)(wrow + k0);
      const v2f a1 = *(const v2f*)(xrow + k0 + 4);
      const v2f b1 = *(const v2f*)(wrow + k0 + 4);
      c0 = __builtin_amdgcn_wmma_f32_16x16x4_f32(false, a0, false, b0,
                                                 (short)0, c0, false, false);
      c1 = __builtin_amdgcn_wmma_f32_16x16x4_f32(false, a1, false, b1,
                                                 (short)0, c1, false, false);
    }

    const float bv = bias[v];
    #pragma unroll
    for (int j = 0; j < 8; ++j) {
      // C/D layout: VGPR j holds (M = j + 8*hi, N = lo); dist for column v
      const float d = fmaf(-2.0f, c0[j] + c1[j], bv);
      if (d < best[j]) { best[j] = d; bidx[j] = v; }  // strict <: v increasing
    }
  }

  // ---- Reduce across the 16 lanes of each half (rows fixed per half) -------
  #pragma unroll
  for (int j = 0; j < 8; ++j) {
    #pragma unroll
    for (int m = 1; m <= 8; m <<= 1) {
      const float ov = __shfl_xor(best[j], m, 32);
      const int   oi = __shfl_xor(bidx[j], m, 32);
      if (ov < best[j] || (ov == best[j] && oi < bidx[j])) {
        best[j] = ov; bidx[j] = oi;
      }
    }
  }
  if (lo == 0) {  // lane 0 -> rows 0..7, lane 16 -> rows 8..15
    #pragma unroll
    for (int j = 0; j < 8; ++j) {
      redV[wave][hi * 8 + j] = best[j];
      redI[wave][hi * 8 + j] = bidx[j];
    }
  }
  __syncthreads();

  // ---- Reduce across the 8 waves (idx tie-break: columns interleaved) ------
  if (tid < 16) {
    float bv = redV[0][tid];
    int   bi = redI[0][tid];
    #pragma unroll
    for (int w8 = 1; w8 < 8; ++w8) {
      const float ov = redV[w8][tid];
      const int   oi = redI[w8][tid];
      if (ov < bv || (ov == bv && oi < bi)) { bv = ov; bi = oi; }
    }
    fidx[tid] = bi;
  }
  __syncthreads();

  // ---- Gather: out[row, :] = W[fidx[row], :]  (float4, coalesced) ----------
  const int row = tid >> 4;
  const int l16 = tid & 15;
  const float4* src = (const float4*)(w + (long)fidx[row] * EMBED);
  float4* dst = (float4*)(out + (r0 + row) * EMBED);
  #pragma unroll
  for (int i = 0; i < 8; ++i) dst[l16 + i * 16] = src[l16 + i * 16];
}

// ---------------------------------------------------------------------------
extern "C" void kernel_launch(void* const* d_in, const int* in_sizes, int n_in,
                              void* d_out, int out_size, void* d_ws, size_t ws_size,
                              hipStream_t stream) {
  const float* x = (const float*)d_in[0];   // (16,64,64,512) f32 -> 65536 x 512
  const float* w = (const float*)d_in[1];   // (512,512) f32
  float* out  = (float*)d_out;              // 65536 x 512 f32
  float* bias = (float*)d_ws;               // 512 floats of scratch

  quantizer_bias_kernel<<<VOCAB / 256, 256, 0, stream>>>(w, bias);

  const int n_rows = in_sizes[0] / EMBED;   // 65536
  quantizer_main_kernel<<<n_rows / 16, 256, 0, stream>>>(x, w, bias, out);
}